// VQLoss_25357486916145
// MI455X (gfx1250) — compile-verified
//
#include <hip/hip_runtime.h>
#include <hip/hip_bf16.h>
#include <math.h>

// Problem constants (from reference): B=2, Q=64, K=512, N=1024, C=256,
// STRIDE=64, n_wav = N*STRIDE-1 = 65535, BETA=0.25.
#define VQ_B   2
#define VQ_Q   64
#define VQ_K   512
#define VQ_N   1024
#define VQ_C   256
#define VQ_NW  65535

typedef __attribute__((ext_vector_type(2))) float v2f;
typedef __attribute__((ext_vector_type(8))) float v8f;

// ---------------------------------------------------------------------------
// Kernel A: per (b, n) compute l2_loss[b,n] = min_k ( Q*e^2 - 2*e*S1 + S2 )
// with S1 = sum_q ze[b,q,n], S2 = sum_q ze[b,q,n]^2, e = emb[k,n].
// S1/S2 are computed with V_WMMA_F32_16X16X4_F32: A = ones(16x4),
// B = 4x16 chunk of ze (resp. ze^2) over a 16-wide n-tile; D rows all hold
// the K-chunk column sums, accumulated over 16 chunks (Q=64).
// One wave (32 threads) per (b, 16-column tile): EXEC is all ones as WMMA
// requires. gridDim.x = B * (N/16) = 128.
// ---------------------------------------------------------------------------
__global__ __launch_bounds__(32)
void vq_l2_wmma_kernel(const float* __restrict__ ze,
                       const float* __restrict__ emb,
                       float* __restrict__ l2_out) {
    const int idx  = blockIdx.x;          // 0..127
    const int b    = idx >> 6;            // 0..1
    const int tile = idx & 63;            // 0..63
    const int n0   = tile * 16;

    const int lane = threadIdx.x;         // 0..31 (wave32)
    const int col  = lane & 15;
    const int half = lane >> 4;           // 0 or 1

    // A = all ones (16x4 f32 -> 2 VGPRs/lane). With A==1 the exact k-slot
    // placement in B is irrelevant to the accumulated sum.
    v2f a;  a.x = 1.0f; a.y = 1.0f;
    v8f c1 = {};   // accumulates S1 chunks
    v8f c2 = {};   // accumulates S2 chunks

    const float* zb = ze + (size_t)b * VQ_Q * VQ_N;
    const int kslot = half * 2;           // lanes 0-15 -> q-slots {0,1}; 16-31 -> {2,3}

    #pragma unroll
    for (int q0 = 0; q0 < VQ_Q; q0 += 4) {
        float z0 = zb[(size_t)(q0 + kslot)     * VQ_N + n0 + col];
        float z1 = zb[(size_t)(q0 + kslot + 1) * VQ_N + n0 + col];
        v2f bm;  bm.x  = z0;      bm.y  = z1;
        v2f bm2; bm2.x = z0 * z0; bm2.y = z1 * z1;
        c1 = __builtin_amdgcn_wmma_f32_16x16x4_f32(false, a, false, bm,
                                                   (short)0, c1, false, false);
        c2 = __builtin_amdgcn_wmma_f32_16x16x4_f32(false, a, false, bm2,
                                                   (short)0, c2, false, false);
    }

    // Every element of D holds the column sum; c[0] at lane L corresponds to
    // column N = L & 15 (rows M=0 / M=8 for the two lane halves).
    const float S1 = c1[0];
    const float S2 = c2[0];

    // min over k: lanes 0-15 cover k in [0,256), lanes 16-31 cover [256,512)
    // for the same column; combine halves with a cross-lane shuffle.
    const int n = n0 + col;
    float best = __builtin_inff();
    const int kbeg = half * (VQ_K / 2);
    #pragma unroll 4
    for (int k = kbeg; k < kbeg + VQ_K / 2; ++k) {
        float e = emb[(size_t)k * VQ_N + n];
        // Q*e^2 - 2*e*S1 + S2
        float v = fmaf((float)VQ_Q * e, e, fmaf(-2.0f * e, S1, S2));
        best = fminf(best, v);
    }
    best = fminf(best, __shfl_xor(best, 16, 32));
    if (half == 0) l2_out[b * VQ_N + n] = best;
}

// ---------------------------------------------------------------------------
// Kernel B: streaming online log-softmax + target gather over quant_pred.
// One thread per (b, t); loop over C=256 classes with coalesced, non-temporal
// b32 loads (134 MB single pass -> HBM roofline ~5.8us) and a software
// prefetch stream (global_prefetch_b8). One exp per element via the branchy
// online-softmax update. Per-block LDS reduction -> partials in workspace.
// grid = (256, B), block = 256.
// ---------------------------------------------------------------------------
__global__ __launch_bounds__(256)
void vq_softmax_ce_kernel(const float* __restrict__ qp,
                          const int* __restrict__ target,
                          const float* __restrict__ l2,
                          float* __restrict__ partials) {
    const int b = blockIdx.y;
    const int t = blockIdx.x * blockDim.x + threadIdx.x;

    float local = 0.0f;
    if (t < VQ_NW) {
        const float* base = qp + (size_t)b * VQ_C * VQ_NW + t;
        const int tgt = target[(size_t)b * VQ_NW + t];

        float m  = -__builtin_inff();
        float s  = 0.0f;
        float xt = 0.0f;

        #pragma unroll 4
        for (int c = 0; c < VQ_C; ++c) {
            const float* p = base + (size_t)c * VQ_NW;
            if (c + 16 < VQ_C)
                __builtin_prefetch(p + (size_t)16 * VQ_NW, 0, 1);
            float x = __builtin_nontemporal_load(p);
            if (c == tgt) xt = x;
            if (x <= m) {
                s += __expf(x - m);
            } else {
                s = fmaf(s, __expf(m - x), 1.0f);
                m = x;
            }
        }
        // log_softmax at target: x_t - m - log(sum exp(x - m))
        const float lp = xt - m - __logf(s);
        // combined VQ terms: (1 + BETA) * l2_loss[b, t/STRIDE]
        local = lp + 1.25f * l2[b * VQ_N + (t >> 6)];
    }

    __shared__ float red[256];
    red[threadIdx.x] = local;
    __syncthreads();
    #pragma unroll
    for (int off = 128; off > 0; off >>= 1) {
        if (threadIdx.x < off) red[threadIdx.x] += red[threadIdx.x + off];
        __syncthreads();
    }
    if (threadIdx.x == 0)
        partials[blockIdx.y * gridDim.x + blockIdx.x] = red[0];
}

// ---------------------------------------------------------------------------
// Kernel C: deterministic single-block finalize (no float atomics so graph
// replays are bit-identical). mean divisor = B * n_wav = 131070.
// ---------------------------------------------------------------------------
__global__ __launch_bounds__(256)
void vq_finalize_kernel(const float* __restrict__ partials, int n,
                        float* __restrict__ out) {
    float v = 0.0f;
    for (int i = threadIdx.x; i < n; i += 256) v += partials[i];
    __shared__ float red[256];
    red[threadIdx.x] = v;
    __syncthreads();
    #pragma unroll
    for (int off = 128; off > 0; off >>= 1) {
        if (threadIdx.x < off) red[threadIdx.x] += red[threadIdx.x + off];
        __syncthreads();
    }
    if (threadIdx.x == 0)
        out[0] = red[0] * (1.0f / (float)(VQ_B * VQ_NW));
}

// ---------------------------------------------------------------------------
extern "C" void kernel_launch(void* const* d_in, const int* in_sizes, int n_in,
                              void* d_out, int out_size, void* d_ws, size_t ws_size,
                              hipStream_t stream) {
    const float* qp  = (const float*)d_in[0];   // (B, C, n_wav) f32
    const float* ze  = (const float*)d_in[1];   // (B, Q, N)     f32
    const float* emb = (const float*)d_in[2];   // (K, N)        f32
    const int*   tgt = (const int*)d_in[3];     // (B, 1, n_wav) i32
    float* out = (float*)d_out;

    float* l2       = (float*)d_ws;             // B*N = 2048 floats
    float* partials = l2 + VQ_B * VQ_N;         // 512 floats

    // A: WMMA-based S1/S2 + min-over-codebook  (128 waves)
    vq_l2_wmma_kernel<<<dim3(VQ_B * (VQ_N / 16)), dim3(32), 0, stream>>>(ze, emb, l2);

    // B: streaming online softmax-CE  (grid 256 x B, 256 threads)
    dim3 grid((VQ_NW + 255) / 256, VQ_B);
    vq_softmax_ce_kernel<<<grid, dim3(256), 0, stream>>>(qp, tgt, l2, partials);

    // C: finalize mean
    const int nPart = (int)(grid.x * grid.y);
    vq_finalize_kernel<<<dim3(1), dim3(256), 0, stream>>>(partials, nPart, out);
}